// LGNAtomLevel_49572512531083
// MI455X (gfx1250) — compile-verified
//
#include <hip/hip_runtime.h>

typedef __attribute__((ext_vector_type(2))) float v2f;
typedef __attribute__((ext_vector_type(8))) float v8f;

#if defined(__has_builtin)
#if __has_builtin(__builtin_amdgcn_tensor_load_to_lds) && __has_builtin(__builtin_amdgcn_s_wait_tensorcnt)
#define USE_TDM 1
#endif
#endif
#ifndef USE_TDM
#define USE_TDM 0
#endif

namespace {
constexpr int B = 8, N = 128, C = 32;
// LDS layout (dwords). Double-buffered raw stage aliased by T (separated by barriers).
constexpr int SM_BUF0  = 0;     // [2304] raw chunk: atoms(1152) + edges(1152)
constexpr int SM_BUF1  = 2304;
constexpr int SM_T     = 0;     // [32][16][16] = 8192, aliases both buffers
constexpr int SM_CAT   = 8192;  // 3360 concatenated features
constexpr int SM_CG    = 11552; // 540 CG coefficients
constexpr int SM_AI    = 12096; // [32][16] atom(b,i) padded
constexpr int SM_MASK  = 12608; // [128] mask row as float
constexpr int SM_TOTAL = 12736; // 50944 bytes
// raw region offsets within one stage buffer (dwords)
constexpr int RA0 = 0, RA1 = 128, RA2 = 512;      // atoms: 128 + 384 + 640
constexpr int RE0 = 1152, RE1 = 1280, RE2 = 1664; // edges: 128 + 384 + 640
}

__device__ const int g_dims[3]     = {1, 3, 5};
__device__ const int g_xoff[3]     = {0, 1, 4};
__device__ const int g_npairs[3]   = {3, 5, 6};
__device__ const int g_pairbase[3] = {0, 3, 8};
// pair slots in PAIRS order: l=0:(00)(11)(22)  l=1:(01)(10)(11)(12)(21)  l=2:(02)(11)(12)(20)(21)(22)
__device__ const int g_pl1[14]   = {0,1,2, 0,1,1,1,2, 0,1,1,2,2,2};
__device__ const int g_pl2[14]   = {0,1,2, 1,0,1,2,1, 2,1,2,0,1,2};
__device__ const int g_cgoff[14] = {0,1,10, 35,44,53,80,125, 170,195,240,315,340,415};
__device__ const int g_cgsz[14]  = {1,9,25, 9,9,27,45,45, 25,45,75,25,75,125};
__device__ const int g_catbase[3]= {0, 224, 1280};

// branch-light wide copy: 288 float4 = one raw 3-tensor chunk (sizes 32/96/160 f4)
__device__ __forceinline__ void copy3(float* dst, const float4* s0, const float4* s1,
                                      const float4* s2, int tid) {
  float4* d = (float4*)dst;
  for (int e = tid; e < 288; e += 256) {
    float4 v;
    if (e < 32)       v = s0[e];
    else if (e < 128) v = s1[e - 32];
    else              v = s2[e - 128];
    d[e] = v;
  }
}

#if USE_TDM
typedef unsigned __attribute__((ext_vector_type(4))) u32x4;
typedef int      __attribute__((ext_vector_type(8))) i32x8;
typedef int      __attribute__((ext_vector_type(4))) i32x4;

// 1-D TDM descriptor per CDNA5 ISA 8.3/8.4: count=1, type=2("image"),
// data_size=4B, tile_dim0 = tensor_dim0 = stride = ndw dwords, 1-D tile.
// amdgpu-toolchain (clang-23) 6-arg builtin form:
//   (uint32x4 g0, int32x8 g1, int32x4 g2, int32x4 g3, int32x8 extra, i32 cpol)
__device__ __forceinline__ void tdm_load_1d(const float* g, const float* ldsp, unsigned ndw) {
  unsigned long long ga = (unsigned long long)g;
  unsigned la = (unsigned)(unsigned long long)ldsp;  // low 32 bits = LDS byte offset
  u32x4 g0;
  g0[0] = 1u;                                             // count=1, user descriptor
  g0[1] = la;                                             // lds_addr
  g0[2] = (unsigned)ga;                                   // global_addr[31:0]
  g0[3] = ((unsigned)(ga >> 32) & 0x01ffffffu) | (2u << 30); // global_addr[56:32] | type=2
  i32x8 g1;
  g1[0] = (int)(2u << 16);                                // data_size = 4 bytes
  g1[1] = (int)((ndw & 0xffffu) << 16);                   // tensor_dim0[15:0]
  g1[2] = (int)((ndw >> 16) & 0xffffu);                   // tensor_dim0[31:16]
  g1[3] = (int)((ndw & 0xffffu) << 16);                   // tile_dim0
  g1[4] = 0;                                              // tile_dim1 = tile_dim2 = 0 (1-D)
  g1[5] = (int)ndw;                                       // tensor_dim0_stride[31:0]
  g1[6] = 0;
  g1[7] = 0;
  i32x4 gz4 = (i32x4)0;
  i32x8 gz8 = (i32x8)0;
  __builtin_amdgcn_tensor_load_to_lds(g0, g1, gz4, gz4, gz8, 0);
}
#endif

__global__ __launch_bounds__(256)
void lgn_atom_level_kernel(
    const float* __restrict__ atom0, const float* __restrict__ edge0,
    const float* __restrict__ atom1, const float* __restrict__ edge1,
    const float* __restrict__ atom2, const float* __restrict__ edge2,
    const unsigned char* __restrict__ mask,
    const float* __restrict__ cg000, const float* __restrict__ cg110, const float* __restrict__ cg220,
    const float* __restrict__ cg011, const float* __restrict__ cg101, const float* __restrict__ cg111,
    const float* __restrict__ cg121, const float* __restrict__ cg211,
    const float* __restrict__ cg022, const float* __restrict__ cg112, const float* __restrict__ cg122,
    const float* __restrict__ cg202, const float* __restrict__ cg212, const float* __restrict__ cg222,
    const float* __restrict__ w0, const float* __restrict__ w1, const float* __restrict__ w2,
    float* __restrict__ out)
{
  __shared__ __attribute__((aligned(16))) float sm[SM_TOTAL];
  const int tid  = threadIdx.x;
  const int bi   = blockIdx.x;
  const int bb   = bi >> 7;
  const int ii   = bi & (N - 1);
  const int lane = tid & 31;        // wave32
  const int wv   = tid >> 5;        // 8 waves/block

  // ---- per-lane gather constants (x/y -> raw region), computed once, no divergence later
  const int y   = lane & 15;
  const int k2  = (lane >> 4) << 1;  // K pair base: lanes 0-15 -> K{0,1}, 16-31 -> K{2,3}
  const int dr  = (y == 0) ? 1 : (y < 4) ? 3 : (y < 9) ? 5 : 0;  // dr=0 -> safe dummy read
  const int yl  = (y == 0) ? 0 : (y < 4) ? y - 1 : (y < 9) ? y - 4 : 0;
  const int rba = (y == 0) ? RA0 : (y < 4) ? RA1 : (y < 9) ? RA2 : RA0;
  const int rbe = (y == 0) ? RE0 : (y < 4) ? RE1 : (y < 9) ? RE2 : RE0;

  // ---- stage CG coefficients, atom(b,i), and mask row once ----
  {
    const float* cgp[14] = {cg000,cg110,cg220, cg011,cg101,cg111,cg121,cg211,
                            cg022,cg112,cg122,cg202,cg212,cg222};
    for (int t = 0; t < 14; ++t)
      for (int e = tid; e < g_cgsz[t]; e += 256)
        sm[SM_CG + g_cgoff[t] + e] = cgp[t][e];
  }
  for (int e = tid; e < C * 16; e += 256) {
    int c = e >> 4, x = e & 15;
    int base = (bb * N + ii) * C + c;
    float v = 0.f;
    if (x == 0)      v = atom0[base];
    else if (x < 4)  v = atom1[base * 3 + (x - 1)];
    else if (x < 9)  v = atom2[base * 5 + (x - 4)];
    sm[SM_AI + e] = v;
  }
  for (int e = tid; e < N; e += 256)
    sm[SM_MASK + e] = (float)mask[(bb * N + ii) * N + e];

  // ---- chunk stager: raw contiguous copies (TDM for edges, b128 for atoms) ----
  auto stage = [&](int bufd, int jb2) {
    long long er = (long long)((bb * N + ii) * N + jb2);
    const float* e0 = edge0 + er * C;
    const float* e1 = edge1 + er * C * 3;
    const float* e2 = edge2 + er * C * 5;
    long long ar = (long long)(bb * N + jb2);
    const float* a0 = atom0 + ar * C;
    const float* a1 = atom1 + ar * C * 3;
    const float* a2 = atom2 + ar * C * 5;
#if USE_TDM
    if (wv == 0) {
      tdm_load_1d(e0, &sm[bufd + RE0], 128);
      tdm_load_1d(e1, &sm[bufd + RE1], 384);
      tdm_load_1d(e2, &sm[bufd + RE2], 640);
    }
#else
    copy3(&sm[bufd + RE0], (const float4*)e0, (const float4*)e1, (const float4*)e2, tid);
#endif
    copy3(&sm[bufd + RA0], (const float4*)a0, (const float4*)a1, (const float4*)a2, tid);
  };

  // ---- main loop: T[c] = sum_j atom(b,j,c,:) (outer) mask*edge(b,i,j,c,:) via WMMA f32 16x16x4
  stage(SM_BUF0, 0);   // preload chunk 0
  v8f acc[4] = {};
  for (int t = 0; t < 32; ++t) {
    const int curd = (t & 1) ? SM_BUF1 : SM_BUF0;
    const int nxtd = (t & 1) ? SM_BUF0 : SM_BUF1;
    __syncthreads();                       // prior reads of nxt buffer complete
    if (t + 1 < 32) stage(nxtd, (t + 1) * 4);
#if USE_TDM
    if (wv == 0) {
      if (t + 1 < 32) __builtin_amdgcn_s_wait_tensorcnt(3);  // oldest 3 (chunk t) done
      else            __builtin_amdgcn_s_wait_tensorcnt(0);
    }
#endif
    __syncthreads();                       // chunk t data visible to all waves
    const int jb = t * 4;
    const float m0 = sm[SM_MASK + jb + k2];
    const float m1 = sm[SM_MASK + jb + k2 + 1];
#pragma unroll
    for (int cc = 0; cc < 4; ++cc) {       // 4 channels per wave, independent accumulators
      const int c  = (wv << 2) + cc;
      const int ja = (k2 * 32 + c) * dr + yl;
      v2f af = { sm[curd + rba + ja], sm[curd + rba + ja + 32 * dr] };
      v2f bf = { sm[curd + rbe + ja] * m0, sm[curd + rbe + ja + 32 * dr] * m1 };
      acc[cc] = __builtin_amdgcn_wmma_f32_16x16x4_f32(
          false, af, false, bf, (short)0, acc[cc], false, false);
    }
  }

  // ---- dump T accumulators to LDS: T[c][x][y] (aliases stage buffers) ----
  __syncthreads();
  {
    int xh = (lane >> 4) << 3;   // C/D layout: vgpr r -> x = r + 8*(lane/16)
#pragma unroll
    for (int cc = 0; cc < 4; ++cc) {
      int c = (wv << 2) + cc;
#pragma unroll
      for (int r = 0; r < 8; ++r)
        sm[SM_T + (c * 16 + xh + r) * 16 + y] = acc[cc][r];
    }
  }
  __syncthreads();

  // ---- build concatenated features: [agg pairs | identity | power pairs] x C, per level ----
  for (int e = tid; e < 3360; e += 256) {
    int l, rem;
    if (e < 224)       { l = 0; rem = e; }
    else if (e < 1280) { l = 1; rem = e - 224; }
    else               { l = 2; rem = e - 1280; }
    int d = g_dims[l];
    int k = rem / d;
    int z = rem - k * d;
    int part = k >> 5;
    int c = k & 31;
    int P = g_npairs[l];
    float val;
    if (part == P) {                       // identity
      val = sm[SM_AI + c * 16 + g_xoff[l] + z];
    } else {
      int p    = (part < P) ? part : (part - P - 1);
      int slot = g_pairbase[l] + p;
      int l1 = g_pl1[slot], l2 = g_pl2[slot];
      int d1 = g_dims[l1],  d2 = g_dims[l2];
      int xb = g_xoff[l1],  yb = g_xoff[l2];
      const float* cgt = &sm[SM_CG + g_cgoff[slot]];
      float s = 0.f;
      if (part < P) {                      // aggregate: contract T with CG
        const float* Tc = &sm[SM_T + c * 256];
        for (int x = 0; x < d1; ++x)
          for (int yy = 0; yy < d2; ++yy)
            s += Tc[(xb + x) * 16 + (yb + yy)] * cgt[(x * d2 + yy) * d + z];
      } else {                             // self power: atom(i) x atom(i) x CG
        for (int x = 0; x < d1; ++x) {
          float ax = sm[SM_AI + c * 16 + xb + x];
          for (int yy = 0; yy < d2; ++yy)
            s += ax * sm[SM_AI + c * 16 + yb + yy] * cgt[(x * d2 + yy) * d + z];
        }
      }
      val = s;
    }
    sm[SM_CAT + e] = val;
  }
  __syncthreads();

  // ---- channel mix: out_l[b,i,o,z] = sum_k cat[k,z] * w_l[k,o] ----
  for (int e = tid; e < 288; e += 256) {
    int l, z, o;
    if (e < 32)       { l = 0; z = 0;              o = e; }
    else if (e < 128) { l = 1; z = (e - 32) >> 5;  o = (e - 32) & 31; }
    else              { l = 2; z = (e - 128) >> 5; o = (e - 128) & 31; }
    int d   = g_dims[l];
    int cin = (2 * g_npairs[l] + 1) * C;
    const float* wl = (l == 0) ? w0 : (l == 1) ? w1 : w2;
    const float* cl = &sm[SM_CAT + g_catbase[l]];
    float s = 0.f;
    for (int k = 0; k < cin; ++k)
      s += cl[k * d + z] * wl[k * C + o];
    int obase = (l == 0) ? 0 : (l == 1) ? (B * N * C) : (4 * B * N * C);
    out[obase + ((bb * N + ii) * C + o) * d + z] = s;
  }
}

extern "C" void kernel_launch(void* const* d_in, const int* in_sizes, int n_in,
                              void* d_out, int out_size, void* d_ws, size_t ws_size,
                              hipStream_t stream) {
  (void)in_sizes; (void)n_in; (void)out_size; (void)d_ws; (void)ws_size;
  // setup_inputs() dict order: atom_0, edge_0, atom_1, edge_1, atom_2, edge_2, mask,
  //   cg_000, cg_110, cg_220, cg_011, cg_101, cg_111, cg_121, cg_211,
  //   cg_022, cg_112, cg_122, cg_202, cg_212, cg_222, w_0, w_1, w_2
  const float* atom0 = (const float*)d_in[0];
  const float* edge0 = (const float*)d_in[1];
  const float* atom1 = (const float*)d_in[2];
  const float* edge1 = (const float*)d_in[3];
  const float* atom2 = (const float*)d_in[4];
  const float* edge2 = (const float*)d_in[5];
  const unsigned char* mask = (const unsigned char*)d_in[6];  // jnp bool -> 1 byte
  const float* cg000 = (const float*)d_in[7];
  const float* cg110 = (const float*)d_in[8];
  const float* cg220 = (const float*)d_in[9];
  const float* cg011 = (const float*)d_in[10];
  const float* cg101 = (const float*)d_in[11];
  const float* cg111 = (const float*)d_in[12];
  const float* cg121 = (const float*)d_in[13];
  const float* cg211 = (const float*)d_in[14];
  const float* cg022 = (const float*)d_in[15];
  const float* cg112 = (const float*)d_in[16];
  const float* cg122 = (const float*)d_in[17];
  const float* cg202 = (const float*)d_in[18];
  const float* cg212 = (const float*)d_in[19];
  const float* cg222 = (const float*)d_in[20];
  const float* w0 = (const float*)d_in[21];
  const float* w1 = (const float*)d_in[22];
  const float* w2 = (const float*)d_in[23];
  float* out = (float*)d_out;

  lgn_atom_level_kernel<<<dim3(B * N), dim3(256), 0, stream>>>(
      atom0, edge0, atom1, edge1, atom2, edge2, mask,
      cg000, cg110, cg220, cg011, cg101, cg111, cg121, cg211,
      cg022, cg112, cg122, cg202, cg212, cg222, w0, w1, w2, out);
}